// UMIR_46832323395938
// MI455X (gfx1250) — compile-verified
//
#include <hip/hip_runtime.h>
#include <math.h>

typedef __attribute__((ext_vector_type(2))) float v2f;
typedef __attribute__((ext_vector_type(8))) float v8f;

#define N_ENTITY 500000
#define N_REL    32
#define DIM      64
#define HOPS     2
#define N_ITEM   10000
#define N_MEM    32
#define BATCH    4096
#define HIST     50

// ---------------------------------------------------------------------------
// Kernel 0: rdot[r] = relation_emb[r] . w_r   (only 32 relations -> tiny LUT)
// ---------------------------------------------------------------------------
__global__ __launch_bounds__(32)
void rdot_kernel(const float* __restrict__ rel_emb,
                 const float* __restrict__ W_w,
                 float* __restrict__ rdot) {
  int r = threadIdx.x;           // 32 lanes = 32 relations
  float s = 0.f;
  const float* row = rel_emb + r * DIM;
  #pragma unroll
  for (int d = 0; d < DIM; ++d) s += row[d] * W_w[DIM + d];
  rdot[r] = s;
}

// ---------------------------------------------------------------------------
// Kernel 1: per block of 16 items (one wave32):
//   logits via v_wmma_f32_16x16x4_f32 chains (A = gathered emb rows, B = w
//   chunk replicated across N), softmax(sigmoid) over 32 mems, pi-weighted
//   tail sum accumulated into acc[item].
// ---------------------------------------------------------------------------
__global__ __launch_bounds__(32)
void acc_kernel(const float* __restrict__ entity_emb,
                const float* __restrict__ W_w,
                const float* __restrict__ W_b,
                const int*   __restrict__ item_ids,
                const int*   __restrict__ heads,
                const int*   __restrict__ relations,
                const int*   __restrict__ tails,
                const float* __restrict__ rdot,
                float*       __restrict__ acc_out) {
  __shared__ float ldsW[3 * DIM];     // w_h | w_r | w_t
  __shared__ float ldsLogit[N_MEM];   // per-item logits (32 mems)
  __shared__ float ldsRdot[N_REL];

  const int lane = threadIdx.x;       // 0..31
  const int o    = lane >> 4;         // half-wave: selects K sub-pair
  const int s    = lane & 15;         // A-row / D-column index
  const int i0   = blockIdx.x * 16;

  for (int k = lane; k < 3 * DIM; k += 32) ldsW[k] = W_w[k];
  ldsRdot[lane] = rdot[lane];
  __syncthreads();

  const float bias = W_b[0];

  // acc starts as entity_emb[item_ids]; lane l holds dims {2l, 2l+1}
  v2f accv[16];
  #pragma unroll
  for (int j = 0; j < 16; ++j) {
    int iid = item_ids[i0 + j];
    accv[j] = ((const v2f*)(entity_emb + (size_t)iid * DIM))[lane];
  }

  for (int h = 0; h < HOPS; ++h) {
    for (int j = 0; j < 16; ++j) {
      const int  item = i0 + j;
      const long base = ((long)h * N_ITEM + item) * N_MEM;

      // ---- logits for 32 mems, two WMMA chains of 16 slots each ----
      for (int half = 0; half < 2; ++half) {
        const int mem = half * 16 + s;            // this lane's A-row slot
        const int hid = heads[base + mem];
        const int tid = tails[base + mem];
        const float* hrow = entity_emb + (size_t)hid * DIM;
        const float* trow = entity_emb + (size_t)tid * DIM;

        v8f c = {};                               // C starts at 0
        #pragma unroll
        for (int ck = 0; ck < 16; ++ck) {         // K = 64 in steps of 4
          const int kb = 4 * ck + 2 * o;          // lane's K sub-pair
          v2f ah = *(const v2f*)(hrow + kb);              // A: head row
          v2f bh = *(const v2f*)(ldsW + kb);              // B: w_h (bcast N)
          c = __builtin_amdgcn_wmma_f32_16x16x4_f32(
                  false, ah, false, bh, (short)0, c, false, false);
          v2f at = *(const v2f*)(trow + kb);              // A: tail row
          v2f bt = *(const v2f*)(ldsW + 2 * DIM + kb);    // B: w_t (bcast N)
          c = __builtin_amdgcn_wmma_f32_16x16x4_f32(
                  false, at, false, bt, (short)0, c, false, false);
        }
        // D[M][N] identical across N; lane with N==0 in each half extracts
        // M = g + 8*o  -> mem = half*16 + g + 8*o
        if (s == 0) {
          #pragma unroll
          for (int g = 0; g < 8; ++g) ldsLogit[half * 16 + g + 8 * o] = c[g];
        }
      }
      __syncthreads();

      // ---- softmax(sigmoid(logits)) across the 32 mems (lane = mem) ----
      const int   ridx  = relations[base + lane];
      const float logit = ldsLogit[lane] + bias + ldsRdot[ridx];
      const float sg    = 1.f / (1.f + __expf(-logit));
      const float e     = __expf(sg);
      float sum = e;
      #pragma unroll
      for (int off = 16; off > 0; off >>= 1) sum += __shfl_xor(sum, off, 32);
      const float pi    = e / sum;
      const int   tid_l = tails[base + lane];

      // ---- acc += sum_m pi[m] * te[m, :] ; lane = dim pair, coalesced ----
      v2f a = accv[j];
      for (int m = 0; m < N_MEM; ++m) {
        const float pim = __shfl(pi, m, 32);
        const int   tm  = __shfl(tid_l, m, 32);
        v2f t2 = ((const v2f*)(entity_emb + (size_t)tm * DIM))[lane];
        a.x += pim * t2.x;
        a.y += pim * t2.y;
      }
      accv[j] = a;
      __syncthreads();   // ldsLogit reused next item
    }
  }

  #pragma unroll
  for (int j = 0; j < 16; ++j)
    ((v2f*)(acc_out + (size_t)(i0 + j) * DIM))[lane] = accv[j];
}

// ---------------------------------------------------------------------------
// Kernel 2: user_emb[b] = sum_{j<50} acc[records_idx[b,j]];
//           out[b] = sigmoid(user_emb . entity_emb[items[b]])
// ---------------------------------------------------------------------------
__global__ __launch_bounds__(32)
void user_kernel(const float* __restrict__ entity_emb,
                 const float* __restrict__ acc,
                 const int*   __restrict__ records_idx,
                 const int*   __restrict__ items,
                 float*       __restrict__ out) {
  const int b    = blockIdx.x;
  const int lane = threadIdx.x;
  float ux = 0.f, uy = 0.f;
  const int* rec = records_idx + (size_t)b * HIST;
  for (int j = 0; j < HIST; ++j) {
    v2f r = ((const v2f*)(acc + (size_t)rec[j] * DIM))[lane];
    ux += r.x; uy += r.y;
  }
  v2f p = ((const v2f*)(entity_emb + (size_t)items[b] * DIM))[lane];
  float dot = ux * p.x + uy * p.y;
  #pragma unroll
  for (int off = 16; off > 0; off >>= 1) dot += __shfl_xor(dot, off, 32);
  if (lane == 0) out[b] = 1.f / (1.f + __expf(-dot));
}

// ---------------------------------------------------------------------------
extern "C" void kernel_launch(void* const* d_in, const int* in_sizes, int n_in,
                              void* d_out, int out_size, void* d_ws, size_t ws_size,
                              hipStream_t stream) {
  (void)in_sizes; (void)n_in; (void)out_size; (void)ws_size;
  const float* entity_emb   = (const float*)d_in[0];
  const float* relation_emb = (const float*)d_in[1];
  const float* W_w          = (const float*)d_in[2];
  const float* W_b          = (const float*)d_in[3];
  const int*   item_ids     = (const int*)d_in[4];
  const int*   heads        = (const int*)d_in[5];
  const int*   relations    = (const int*)d_in[6];
  const int*   tails        = (const int*)d_in[7];
  const int*   records_idx  = (const int*)d_in[8];
  const int*   items        = (const int*)d_in[9];
  float*       out          = (float*)d_out;

  float* rdot = (float*)d_ws;        // 32 floats (keeps 128B alignment after)
  float* acc  = rdot + 32;           // N_ITEM * DIM floats (~2.56 MB)

  rdot_kernel<<<1, 32, 0, stream>>>(relation_emb, W_w, rdot);
  acc_kernel<<<N_ITEM / 16, 32, 0, stream>>>(entity_emb, W_w, W_b, item_ids,
                                             heads, relations, tails, rdot, acc);
  user_kernel<<<BATCH, 32, 0, stream>>>(entity_emb, acc, records_idx, items, out);
}